// HGT_30193620091086
// MI455X (gfx1250) — compile-verified
//
#include <hip/hip_runtime.h>

// Problem constants (match reference setup_inputs)
#define NC_N   100000
#define NPR_N  50000
#define E1_N   300000
#define E2_N   300000

typedef __attribute__((ext_vector_type(2))) float v2f;
typedef __attribute__((ext_vector_type(8))) float v8f;

// ---------------- device helpers ----------------
__device__ __forceinline__ float gelu_f(float x) {
  return 0.5f * x * (1.0f + erff(x * 0.70710678118654752f));
}
// order-preserving float<->uint encoding for atomicMax-based segment max
__device__ __forceinline__ unsigned int enc_f(float f) {
  unsigned int u = __float_as_uint(f);
  return (u & 0x80000000u) ? ~u : (u | 0x80000000u);
}
__device__ __forceinline__ float dec_f(unsigned int k) {
  unsigned int u = (k & 0x80000000u) ? (k & 0x7fffffffu) : ~k;
  return __uint_as_float(u);
}

// ---------------- weight folding ----------------
// weff[:,0:128]   = w_k @ blockdiag(a[0],a[1])   (relation key transform folded in)
// weff[:,128:256] = w_q
// weff[:,256:384] = w_v @ blockdiag(m[0],m[1])   (relation value transform folded in)
__global__ void fold_weights(const float* __restrict__ w, const float* __restrict__ b,
                             const float* __restrict__ a, const float* __restrict__ m,
                             float* __restrict__ weff, float* __restrict__ beff) {
  int j = blockIdx.x;      // 0..383 output column
  int i = threadIdx.x;     // 0..127 input row
  float acc, bacc = 0.f;
  if (j >= 128 && j < 256) {
    acc = w[i * 384 + j];
    bacc = b[j];
  } else {
    int base = (j < 128) ? 0 : 256;
    const float* T = (j < 128) ? a : m;   // [H=2, 64, 64]
    int jj = j - base;
    int h = jj >> 6, e = jj & 63;
    acc = 0.f;
    for (int d = 0; d < 64; ++d) {
      float t = T[(h * 64 + d) * 64 + e];
      acc  += w[i * 384 + base + h * 64 + d] * t;
      bacc += b[base + h * 64 + d] * t;
    }
  }
  weff[i * 384 + j] = acc;
  if (i == 0) beff[j] = bacc;
}

// ---------------- WMMA fp32 GEMM ----------------
// C[M,N] = epilogue( act(A[M,K]) @ W[K,N] + bias )
// mode 0: C = acc + bias
// mode 1: o = acc + bias; g = sigmoid(gate[0]); C = g*o + (1-g)*Xres  (+optional relu)
// gelu_a: apply exact GELU to A elements while staging into LDS.
#define BM 128
#define BN 64
#define BK 32
#define LDA 36   // padded LDS strides -> conflict-free, 8B-aligned v2f fragment reads
#define LDB 34

__global__ __launch_bounds__(256) void gemm_wmma(
    const float* __restrict__ A, const float* __restrict__ W,
    const float* __restrict__ bias, const float* __restrict__ Xres,
    const float* __restrict__ gate, float* __restrict__ C,
    int M, int N, int K, int mode, int gelu_a, int relu_out)
{
  __shared__ float sA[BM * LDA];
  __shared__ float sB[BN * LDB];   // W tile stored transposed: sB[n][k]
  const int tid  = threadIdx.x;
  const int wave = tid >> 5;       // 8 waves, each owns 16 rows x 64 cols
  const int lane = tid & 31;
  const int hf   = lane >> 4;      // half-wave
  const int li   = lane & 15;
  const int m0 = blockIdx.x * BM;
  const int n0 = blockIdx.y * BN;

  v8f acc[4];
  for (int t = 0; t < 4; ++t)
    for (int r = 0; r < 8; ++r) acc[t][r] = 0.f;

  for (int k0 = 0; k0 < K; k0 += BK) {
    // stage A tile (128x32) with optional fused GELU
    for (int i = tid; i < (BM * BK) / 4; i += 256) {
      int row = i >> 3;
      int c4  = (i & 7) << 2;
      float4 v = make_float4(0.f, 0.f, 0.f, 0.f);
      int gm = m0 + row;
      if (gm < M) v = *(const float4*)(A + (size_t)gm * K + k0 + c4);
      if (gelu_a) { v.x = gelu_f(v.x); v.y = gelu_f(v.y); v.z = gelu_f(v.z); v.w = gelu_f(v.w); }
      *(float4*)(&sA[row * LDA + c4]) = v;
    }
    // stage W tile (32x64) transposed, N guarded (final layer N=16)
    for (int i = tid; i < BK * BN; i += 256) {
      int r = i >> 6;          // k within tile
      int c = i & 63;          // n within tile
      int gn = n0 + c;
      sB[c * LDB + r] = (gn < N) ? W[(size_t)(k0 + r) * N + gn] : 0.f;
    }
    __syncthreads();

    const int arow = wave * 16 + li;
    #pragma unroll
    for (int kk = 0; kk < BK; kk += 4) {
      // A 16x4 fragment: lane(li,hf) holds A[M=li, K=kk+2*hf .. +1]
      v2f afrag = *(const v2f*)(&sA[arow * LDA + kk + 2 * hf]);
      #pragma unroll
      for (int t = 0; t < 4; ++t) {
        // B 4x16 fragment: lane(li,hf) holds B[K=kk+2*hf .. +1, N=li]
        v2f bfrag = *(const v2f*)(&sB[(t * 16 + li) * LDB + kk + 2 * hf]);
        acc[t] = __builtin_amdgcn_wmma_f32_16x16x4_f32(
            false, afrag, false, bfrag, (short)0, acc[t], false, false);
      }
    }
    __syncthreads();
  }

  float g = 0.f;
  if (mode == 1) g = 1.f / (1.f + expf(-gate[0]));
  #pragma unroll
  for (int t = 0; t < 4; ++t) {
    int n = n0 + t * 16 + li;
    if (n >= N) continue;
    float bval = bias[n];
    #pragma unroll
    for (int r = 0; r < 8; ++r) {
      int m = m0 + wave * 16 + hf * 8 + r;   // C/D: VGPR r -> M = 8*hf + r
      if (m >= M) continue;
      float o = acc[t][r] + bval;
      if (mode == 1) {
        float x = Xres[(size_t)m * N + n];
        o = g * o + (1.f - g) * x;
        if (relu_out) o = fmaxf(o, 0.f);
      }
      C[(size_t)m * N + n] = o;
    }
  }
}

// ---------------- edge kernels ----------------
// h layout per node: [k_rel(0:128) | q(128:256) | v_rel(256:384)]
// merged dst index space: customers [0,NC), products [NC,NC+NPR)

__global__ __launch_bounds__(256) void edge_score(
    const float* __restrict__ hc, const float* __restrict__ hp,
    const int* __restrict__ ecp_s, const int* __restrict__ ecp_d,
    const int* __restrict__ epc_s, const int* __restrict__ epc_d,
    const float* __restrict__ pcp, const float* __restrict__ ppc,
    float* __restrict__ scores, unsigned int* __restrict__ mx)
{
  int e = (blockIdx.x * blockDim.x + threadIdx.x) >> 5;   // wave per edge
  if (e >= E1_N + E2_N) return;
  int lane = threadIdx.x & 31;
  int hf = lane >> 4, li = lane & 15;                     // half-wave per head
  int src, dst; const float* hsrc; float pr;
  if (e < E1_N) { src = ecp_s[e]; dst = ecp_d[e] + NC_N; hsrc = hc; pr = pcp[hf]; }
  else { int t = e - E1_N; src = epc_s[t]; dst = epc_d[t]; hsrc = hp; pr = ppc[hf]; }
  const float* qp = (dst < NC_N) ? (hc + (size_t)dst * 384 + 128)
                                 : (hp + (size_t)(dst - NC_N) * 384 + 128);
  const float* kp = hsrc + (size_t)src * 384;
  int off = hf * 64 + li * 4;
  float4 qv = *(const float4*)(qp + off);
  float4 kv = *(const float4*)(kp + off);
  float p = qv.x * kv.x + qv.y * kv.y + qv.z * kv.z + qv.w * kv.w;
  p += __shfl_xor(p, 1, 32);
  p += __shfl_xor(p, 2, 32);
  p += __shfl_xor(p, 4, 32);
  p += __shfl_xor(p, 8, 32);
  if (li == 0) {
    float s = p * pr * 0.125f;                            // / sqrt(64)
    scores[(size_t)e * 2 + hf] = s;
    atomicMax(mx + (size_t)dst * 2 + hf, enc_f(s));
  }
}

__global__ void edge_exp(const int* __restrict__ ecp_d, const int* __restrict__ epc_d,
                         float* __restrict__ scores, const unsigned int* __restrict__ mx,
                         float* __restrict__ sm)
{
  int idx = blockIdx.x * blockDim.x + threadIdx.x;        // edge*2 + head
  if (idx >= (E1_N + E2_N) * 2) return;
  int e = idx >> 1, h = idx & 1;
  int dst = (e < E1_N) ? (ecp_d[e] + NC_N) : epc_d[e - E1_N];
  float m = dec_f(mx[(size_t)dst * 2 + h]);
  if (!__builtin_isfinite(m)) m = 0.f;                    // matches reference
  float ev = expf(scores[idx] - m);
  scores[idx] = ev;                                       // overwrite with exp value
  atomicAdd(sm + (size_t)dst * 2 + h, ev);
}

__global__ __launch_bounds__(256) void edge_scatter(
    const float* __restrict__ hc, const float* __restrict__ hp,
    const int* __restrict__ ecp_s, const int* __restrict__ ecp_d,
    const int* __restrict__ epc_s, const int* __restrict__ epc_d,
    const float* __restrict__ scores, const float* __restrict__ sm,
    float* __restrict__ agg)
{
  int e = (blockIdx.x * blockDim.x + threadIdx.x) >> 5;
  if (e >= E1_N + E2_N) return;
  int lane = threadIdx.x & 31;
  int hf = lane >> 4, li = lane & 15;
  int src, dst; const float* hsrc;
  if (e < E1_N) { src = ecp_s[e]; dst = ecp_d[e] + NC_N; hsrc = hc; }
  else { int t = e - E1_N; src = epc_s[t]; dst = epc_d[t]; hsrc = hp; }
  float w = scores[(size_t)e * 2 + hf] / (sm[(size_t)dst * 2 + hf] + 1e-16f);
  const float* vp = hsrc + (size_t)src * 384 + 256 + hf * 64 + li * 4;
  float4 vv = *(const float4*)vp;
  float* o = agg + (size_t)dst * 128 + hf * 64 + li * 4;  // 77MB, L2-resident
  atomicAdd(o + 0, w * vv.x);
  atomicAdd(o + 1, w * vv.y);
  atomicAdd(o + 2, w * vv.z);
  atomicAdd(o + 3, w * vv.w);
}

// ---------------- host side ----------------
static inline void launch_gemm(const float* A, const float* W, const float* bias,
                               const float* X, const float* gate, float* C,
                               int M, int N, int mode, int gelu_a, int relu,
                               hipStream_t s)
{
  dim3 grid((M + BM - 1) / BM, (N + BN - 1) / BN);
  gemm_wmma<<<grid, 256, 0, s>>>(A, W, bias, X, gate, C, M, N, 128, mode, gelu_a, relu);
}

extern "C" void kernel_launch(void* const* d_in, const int* in_sizes, int n_in,
                              void* d_out, int out_size, void* d_ws, size_t ws_size,
                              hipStream_t stream) {
  (void)in_sizes; (void)n_in; (void)out_size; (void)ws_size;
  #define FP(i) ((const float*)d_in[i])
  #define IP(i) ((const int*)d_in[i])
  const float* xc = FP(0);
  const float* xp = FP(1);
  // layer param index base: layer1 -> 2, layer2 -> 18
  const int* ecp_s = IP(36); const int* ecp_d = IP(37);
  const int* epc_s = IP(38); const int* epc_d = IP(39);
  const float* wl = FP(34); const float* bl = FP(35);

  // workspace layout (floats); ~443 MB total
  float* ws = (float*)d_ws;
  size_t off = 0;
  float* hc   = ws + off; off += (size_t)NC_N * 384;
  float* hp   = ws + off; off += (size_t)NPR_N * 384;
  float* agg  = ws + off; off += (size_t)(NC_N + NPR_N) * 128;
  float* scb  = ws + off; off += (size_t)(E1_N + E2_N) * 2;
  unsigned int* mxb = (unsigned int*)(ws + off); off += (size_t)(NC_N + NPR_N) * 2;
  float* smb  = ws + off; off += (size_t)(NC_N + NPR_N) * 2;
  float* o1c  = ws + off; off += (size_t)NC_N * 128;
  float* o1p  = ws + off; off += (size_t)NPR_N * 128;
  float* o2c  = ws + off; off += (size_t)NC_N * 128;
  float* wec  = ws + off; off += 128 * 384;
  float* wep  = ws + off; off += 128 * 384;
  float* bec  = ws + off; off += 384;
  float* bep  = ws + off; off += 384;

  const int eblocks  = (E1_N + E2_N + 7) / 8;              // 8 waves/block, wave/edge
  const int e2blocks = ((E1_N + E2_N) * 2 + 255) / 256;

  for (int l = 0; l < 2; ++l) {
    int pb = 2 + l * 16;   // per-layer param base
    const float* Xc = (l == 0) ? xc : o1c;
    const float* Xp = (l == 0) ? xp : o1p;

    // 1) fold relation transforms into projection weights
    fold_weights<<<384, 128, 0, stream>>>(FP(pb+0), FP(pb+1), FP(pb+4), FP(pb+5), wec, bec);
    fold_weights<<<384, 128, 0, stream>>>(FP(pb+2), FP(pb+3), FP(pb+7), FP(pb+8), wep, bep);

    // 2) fused KQV+relation GEMMs -> h = [k_rel | q | v_rel]
    launch_gemm(Xc, wec, bec, nullptr, nullptr, hc, NC_N, 384, 0, 0, 0, stream);
    launch_gemm(Xp, wep, bep, nullptr, nullptr, hp, NPR_N, 384, 0, 0, 0, stream);

    // 3) segment-softmax scratch init
    hipMemsetAsync(mxb, 0, (size_t)(NC_N + NPR_N) * 2 * sizeof(unsigned int), stream);
    hipMemsetAsync(smb, 0, (size_t)(NC_N + NPR_N) * 2 * sizeof(float), stream);
    hipMemsetAsync(agg, 0, (size_t)(NC_N + NPR_N) * 128 * sizeof(float), stream);

    // 4) attention: scores+max, exp+sum, weighted scatter
    edge_score<<<eblocks, 256, 0, stream>>>(hc, hp, ecp_s, ecp_d, epc_s, epc_d,
                                            FP(pb+6), FP(pb+9), scb, mxb);
    edge_exp<<<e2blocks, 256, 0, stream>>>(ecp_d, epc_d, scb, mxb, smb);
    edge_scatter<<<eblocks, 256, 0, stream>>>(hc, hp, ecp_s, ecp_d, epc_s, epc_d,
                                              scb, smb, agg);

    // 5) output mix: gelu(agg) @ ow + ob, sigmoid-gated residual (+relu after layer 1)
    if (l == 0) {
      launch_gemm(agg, FP(pb+10), FP(pb+11), Xc, FP(pb+14), o1c, NC_N, 128, 1, 1, 1, stream);
      launch_gemm(agg + (size_t)NC_N * 128, FP(pb+12), FP(pb+13), Xp, FP(pb+15), o1p,
                  NPR_N, 128, 1, 1, 1, stream);
    } else {
      // layer-2 product mix is dead code w.r.t. the final output -> skipped
      launch_gemm(agg, FP(pb+10), FP(pb+11), Xc, FP(pb+14), o2c, NC_N, 128, 1, 1, 0, stream);
    }
  }

  // final linear: out = o2c @ wl + bl  (N=16, guarded inside the GEMM)
  launch_gemm(o2c, wl, bl, nullptr, nullptr, (float*)d_out, NC_N, 16, 0, 0, 0, stream);
  #undef FP
  #undef IP
}